// DetectionLayer_8624294330475
// MI455X (gfx1250) — compile-verified
//
#include <hip/hip_runtime.h>
#include <hip/hip_bf16.h>

#define BB 16
#define NN 6000
#define GG 512
#define THREADS 256

// ---- gfx1250 async global->LDS copy availability (device pass only) ----
#if defined(__HIP_DEVICE_COMPILE__) && defined(__gfx1250__)
#  if __has_builtin(__builtin_amdgcn_global_load_async_to_lds_b128)
#    define USE_ASYNC_LDS 1
#  endif
#endif

// Builtin parameter types (from hipcc diagnostic): pointers to GCC-vector int4
// in AS1 (global) and AS3 (LDS).
typedef int v4i __attribute__((vector_size(16)));
typedef __attribute__((address_space(1))) v4i gv4i; // global int4
typedef __attribute__((address_space(3))) v4i sv4i; // LDS int4

__device__ __forceinline__ gv4i* as_global_b128(const void* p) {
    // generic global address == AS1 address; integer round-trip avoids
    // clang's cross-AS pointer-cast restriction
    return (gv4i*)(unsigned long long)p;
}
__device__ __forceinline__ sv4i* as_shared_b128(const void* p) {
    // low 32 bits of a generic LDS pointer are the LDS offset (aperture rule)
    return (sv4i*)(unsigned int)(unsigned long long)p;
}

__global__ void __launch_bounds__(THREADS)
detection_layer_kernel(const float* __restrict__ rois,      // (B,N,4)
                       const int*   __restrict__ gt_ids,    // (B,G)
                       const float* __restrict__ gt_boxes,  // (B,G,4)
                       float*       __restrict__ out)
{
    __shared__ float4 s_box[GG];   // gt boxes, AoS (16B-aligned -> ds_load_b128)
    __shared__ float2 s_meta[GG];  // x = gt area, y = flag (0=invalid,1=crowd,2=non-crowd)

    const int tid = threadIdx.x;
    const int b   = blockIdx.y;
    const int n   = blockIdx.x * THREADS + tid;

    const float* gtb = gt_boxes + (size_t)b * GG * 4;

    // ---- Stage 1: gt boxes (8 KB) global -> LDS ----
#if USE_ASYNC_LDS
    // Each thread async-copies two 16B boxes straight into LDS (no VGPR round-trip).
    __builtin_amdgcn_global_load_async_to_lds_b128(
        as_global_b128(gtb + (size_t)tid * 4),
        as_shared_b128(&s_box[tid]), 0, 0);
    __builtin_amdgcn_global_load_async_to_lds_b128(
        as_global_b128(gtb + (size_t)(tid + THREADS) * 4),
        as_shared_b128(&s_box[tid + THREADS]), 0, 0);
#  if __has_builtin(__builtin_amdgcn_s_wait_asynccnt)
    __builtin_amdgcn_s_wait_asynccnt(0);
#  else
    asm volatile("s_wait_asynccnt 0" ::: "memory");
#  endif
#else
    {
        const float4* gtb4 = (const float4*)gtb;
        s_box[tid]           = gtb4[tid];
        s_box[tid + THREADS] = gtb4[tid + THREADS];
    }
#endif
    __syncthreads();

    // ---- Stage 2: per-gt area + crowd/non-crowd flag (computed once per block) ----
    {
        const int* gids = gt_ids + (size_t)b * GG;
        #pragma unroll
        for (int g = tid; g < GG; g += THREADS) {
            float4 q = s_box[g];
            int   id = gids[g];
            bool valid = (fabsf(q.x) + fabsf(q.y) + fabsf(q.z) + fabsf(q.w)) > 0.0f;
            float flag = valid ? ((id < 0) ? 1.0f : ((id > 0) ? 2.0f : 0.0f)) : 0.0f;
            float area = (q.z - q.x) * (q.w - q.y);
            s_meta[g] = make_float2(area, flag);
        }
    }
    __syncthreads();

    // ---- Stage 3: one ROI per thread, sweep all 512 gt via LDS broadcast ----
    float4 r = make_float4(0.0f, 0.0f, 0.0f, 0.0f);
    const bool in_range = (n < NN);
    if (in_range) r = ((const float4*)rois)[(size_t)b * NN + n];

    const bool  valid_roi = (fabsf(r.x) + fabsf(r.y) + fabsf(r.z) + fabsf(r.w)) > 0.0f;
    const float a1 = (r.z - r.x) * (r.w - r.y);

    float ncmax = 0.0f;  // max IoU over non-crowd gt
    float cmax  = 0.0f;  // max IoU over crowd gt

    #pragma unroll 8
    for (int g = 0; g < GG; ++g) {
        float4 q = s_box[g];   // ds_load_b128, same addr in all lanes -> broadcast
        float2 m = s_meta[g];  // ds_load_b64 broadcast
        float y1 = fmaxf(r.x, q.x);
        float x1 = fmaxf(r.y, q.y);
        float y2 = fminf(r.z, q.z);
        float x2 = fminf(r.w, q.w);
        float ih = fmaxf(y2 - y1, 0.0f);
        float iw = fmaxf(x2 - x1, 0.0f);
        float inter = ih * iw;
        float uni   = a1 + m.x - inter;
        float iou   = inter * __builtin_amdgcn_rcpf(uni + 1e-8f); // v_rcp_f32
        ncmax = fmaxf(ncmax, (m.y == 2.0f) ? iou : 0.0f);
        cmax  = fmaxf(cmax,  (m.y == 1.0f) ? iou : 0.0f);
    }

    if (!valid_roi) { ncmax = 0.0f; cmax = 0.0f; }

    if (in_range) {
        // iou_maxes: (B, 2, N)
        float* iou_out = out;
        iou_out[((size_t)b * 2 + 0) * NN + n] = ncmax;
        iou_out[((size_t)b * 2 + 1) * NN + n] = cmax;
        // positive_mask: (B, N) ; negative_mask: (B, N)
        float* posp = out + (size_t)BB * 2 * NN;
        float* negp = posp + (size_t)BB * NN;
        bool pos = valid_roi && (ncmax >= 0.5f);
        bool neg = valid_roi && (ncmax < 0.5f) && (cmax < 0.001f);
        posp[(size_t)b * NN + n] = pos ? 1.0f : 0.0f;
        negp[(size_t)b * NN + n] = neg ? 1.0f : 0.0f;
    }
}

extern "C" void kernel_launch(void* const* d_in, const int* in_sizes, int n_in,
                              void* d_out, int out_size, void* d_ws, size_t ws_size,
                              hipStream_t stream) {
    (void)in_sizes; (void)n_in; (void)out_size; (void)d_ws; (void)ws_size;
    const float* rois     = (const float*)d_in[0];
    const int*   gt_ids   = (const int*)  d_in[1];
    const float* gt_boxes = (const float*)d_in[2];
    float*       out      = (float*)d_out;

    dim3 grid((NN + THREADS - 1) / THREADS, BB);
    detection_layer_kernel<<<grid, dim3(THREADS), 0, stream>>>(rois, gt_ids, gt_boxes, out);
}